// DeformableConv2d_41618233099165
// MI455X (gfx1250) — compile-verified
//
#include <hip/hip_runtime.h>

// Deformable conv for MI455X (gfx1250, wave32).
// Kernel 0: weights -> K-pair-interleaved layout wt2[t][c>>1][o*2+(c&1)].
// Kernel 1: offset conv (64->18ch, 3x3) with weights in LDS.
// Kernel 2: fused bilinear-gather + fp32 WMMA GEMM; both operands stored in
//           LDS in WMMA-operand order so each operand is one ds_load_b64.

typedef __attribute__((ext_vector_type(2))) float v2f;
typedef __attribute__((ext_vector_type(8))) float v8f;

#define B_    4
#define C_    64
#define O_    64
#define H_    128
#define W_    128
#define HW_   16384          // H_*W_
#define NOFF  18             // 2 * 3 * 3 offset channels
#define KPT   9              // taps
#define BROW  160            // B slab row stride (words): 128 data + 32 pad
#define EPAD  17             // padded row stride for epilogue slab

__device__ __forceinline__ int iclamp(int v, int lo, int hi) {
    return v < lo ? lo : (v > hi ? hi : v);
}

// ---------------- Kernel 0: weight transpose + K-pair interleave ----------
// wt2[t*4096 + (c>>1)*128 + o*2 + (c&1)] = w[o*576 + c*9 + t]
__global__ __launch_bounds__(256) void transpose_w_kernel(
    const float* __restrict__ w, float* __restrict__ wt)
{
    const int i = blockIdx.x * 256 + threadIdx.x;   // 0..36863
    const int o = i & 63;
    const int c = (i >> 6) & 63;
    const int t = i >> 12;                          // 0..8
    wt[t * 4096 + (c >> 1) * 128 + o * 2 + (c & 1)] = w[o * 576 + c * 9 + t];
}

// ---------------- Kernel 1: offset convolution ----------------
__global__ __launch_bounds__(256) void offset_conv_kernel(
    const float* __restrict__ x, const float* __restrict__ w_off,
    const float* __restrict__ b_off, float* __restrict__ offs)
{
    __shared__ float wl[NOFF * C_ * 9];   // 41.5 KB
    for (int i = threadIdx.x; i < NOFF * C_ * 9; i += 256) wl[i] = w_off[i];
    __syncthreads();

    const int p   = blockIdx.x * 256 + threadIdx.x;   // linear pixel
    const int b   = p >> 14;
    const int rem = p & (HW_ - 1);
    const int h   = rem >> 7;
    const int w   = rem & (W_ - 1);

    float acc[NOFF];
#pragma unroll
    for (int ch = 0; ch < NOFF; ++ch) acc[ch] = b_off[ch];

    const float* xb = x + b * (C_ * HW_);
    for (int c = 0; c < C_; ++c) {
        const float* xc = xb + c * HW_;
        float patch[9];
#pragma unroll
        for (int i = 0; i < 3; ++i) {
            const int y   = h + i - 1;
            const int ycl = iclamp(y, 0, H_ - 1);
            const bool yok = (y >= 0) & (y < H_);
#pragma unroll
            for (int j = 0; j < 3; ++j) {
                const int xx  = w + j - 1;
                const int xcl = iclamp(xx, 0, W_ - 1);
                const bool ok = yok & (xx >= 0) & (xx < W_);
                const float v = xc[ycl * W_ + xcl];       // clamped: never OOB
                patch[i * 3 + j] = ok ? v : 0.0f;         // zero-pad semantics
            }
        }
#pragma unroll
        for (int ch = 0; ch < NOFF; ++ch) {
            const float* wp = &wl[ch * (C_ * 9) + c * 9];
            float a = acc[ch];
#pragma unroll
            for (int t = 0; t < 9; ++t) a += patch[t] * wp[t];
            acc[ch] = a;
        }
    }
    float* ob = offs + b * (NOFF * HW_) + h * W_ + w;
#pragma unroll
    for (int ch = 0; ch < NOFF; ++ch) ob[ch * HW_] = acc[ch];
}

// ---------- Kernel 2: fused bilinear gather + WMMA GEMM ----------
// One wave per 16-pixel tile (16 consecutive w in one (b,h) row);
// 8 waves (256 threads) per block. All operands flow through LDS in
// WMMA-operand order: one ds_load_b64 per operand, zero repacking.
__global__ __launch_bounds__(256) void deform_wmma_kernel(
    const float* __restrict__ x, const float* __restrict__ offs,
    const float* __restrict__ wt, const float* __restrict__ bias,
    float* __restrict__ out)
{
    __shared__ float swtp[32 * BROW];        // B slab: 32 pair-rows, 20 KB
    __shared__ float slab[8][C_ * EPAD];     // per-wave: A slab / epilogue slab

    const int tid  = threadIdx.x;
    const int wid  = tid >> 5;
    const int lane = tid & 31;
    const int half = lane >> 4;              // 0: lanes 0-15, 1: lanes 16-31
    const int ml   = lane & 15;              // pixel-in-tile / N index
    float* sw = &slab[wid][0];

    const int pbase = (blockIdx.x * 8 + wid) * 16;   // tile = one (b,h) row chunk
    const int b     = pbase >> 14;
    const int rem   = pbase & (HW_ - 1);
    const int h     = rem >> 7;
    const int wb    = rem & (W_ - 1);        // multiple of 16
    const int wpx   = wb + ml;               // this lane's pixel column

    v8f acc0 = {}, acc1 = {}, acc2 = {}, acc3 = {};

    const float* xb   = x + b * (C_ * HW_);
    const float* offp = offs + b * (NOFF * HW_) + h * W_ + wpx;

    for (int t = 0; t < KPT; ++t) {
        // ---- cooperative stage of interleaved weight slice into LDS ----
        const float* wslice = wt + t * (C_ * O_);
#pragma unroll
        for (int j = 0; j < 4; ++j) {
            const int i4  = (tid + j * 256) * 4;     // 0..4092, step 4
            const int pr  = i4 >> 7;                 // pair-row 0..31
            const int rw  = i4 & 127;                // word within row
            const float4 v4 = *(const float4*)(wslice + i4);
            *(float4*)(&swtp[pr * BROW + rw]) = v4;  // 16B aligned
        }
        if (t < KPT - 1)
            __builtin_prefetch(wt + (t + 1) * (C_ * O_) + tid * 16, 0, 1);

        // ---- per-pixel bilinear parameters (uniform control flow) ----
        const float dy = offp[(2 * t)     * HW_];
        const float dx = offp[(2 * t + 1) * HW_];
        const float py = (float)(h   + (t / 3) - 1) + dy;
        const float px = (float)(wpx + (t % 3) - 1) + dx;
        const float y0f = floorf(py), x0f = floorf(px);
        const float fy = py - y0f,    fx = px - x0f;
        const int y0 = (int)y0f, x0 = (int)x0f;
        const int y1 = y0 + 1,   x1 = x0 + 1;
        const int y0c = iclamp(y0, 0, H_ - 1), x0c = iclamp(x0, 0, W_ - 1);
        const int y1c = iclamp(y1, 0, H_ - 1), x1c = iclamp(x1, 0, W_ - 1);
        const bool y0v = (y0 >= 0) & (y0 < H_);
        const bool y1v = (y1 >= 0) & (y1 < H_);
        const bool x0v = (x0 >= 0) & (x0 < W_);
        const bool x1v = (x1 >= 0) & (x1 < W_);
        const float w00 = (y0v & x0v) ? (1.f - fy) * (1.f - fx) : 0.f;
        const float w01 = (y0v & x1v) ? (1.f - fy) * fx         : 0.f;
        const float w10 = (y1v & x0v) ? fy * (1.f - fx)         : 0.f;
        const float w11 = (y1v & x1v) ? fy * fx                 : 0.f;
        const int i00 = y0c * W_ + x0c, i01 = y0c * W_ + x1c;
        const int i10 = y1c * W_ + x0c, i11 = y1c * W_ + x1c;

        // ---- stage s into this wave's slab, K-pair interleaved ----
        // c = 2*it + half  ->  word (c>>1)*32 + 2*m + (c&1)
#pragma unroll 4
        for (int it = 0; it < 32; ++it) {
            const int c = it * 2 + half;
            const float* xc = xb + c * HW_;
            const float s = xc[i00] * w00 + xc[i01] * w01 +
                            xc[i10] * w10 + xc[i11] * w11;
            sw[it * 32 + 2 * ml + half] = s;         // 32 consecutive words/iter
        }
        __syncthreads();   // swtp + all wave slabs ready

        // ---- 16 K-steps: 1 A b64 + 4 B b64 + 4 WMMA each ----
#pragma unroll
        for (int kk = 0; kk < 16; ++kk) {
            const int pr = kk * 2 + half;            // this lane's K-pair row
            const v2f a = *(const v2f*)(&sw[pr * 32 + 2 * ml]);
            const float* bb = &swtp[pr * BROW + 2 * ml];
            const v2f b0 = *(const v2f*)(bb + 0);    // N-tile 0: o = 0..15
            const v2f b1 = *(const v2f*)(bb + 32);   // N-tile 1
            const v2f b2 = *(const v2f*)(bb + 64);   // N-tile 2
            const v2f b3 = *(const v2f*)(bb + 96);   // N-tile 3
            acc0 = __builtin_amdgcn_wmma_f32_16x16x4_f32(false, a, false, b0, (short)0, acc0, false, false);
            acc1 = __builtin_amdgcn_wmma_f32_16x16x4_f32(false, a, false, b1, (short)0, acc1, false, false);
            acc2 = __builtin_amdgcn_wmma_f32_16x16x4_f32(false, a, false, b2, (short)0, acc2, false, false);
            acc3 = __builtin_amdgcn_wmma_f32_16x16x4_f32(false, a, false, b3, (short)0, acc3, false, false);
        }
        __syncthreads();   // protect swtp / slab before next tap overwrites
    }

    // ---- epilogue: transpose through LDS for coalesced stores ----
    // D layout: VGPR r -> M = r + 8*half, N = ml. Write slab as [o*EPAD + m].
#pragma unroll
    for (int r = 0; r < 8; ++r) {
        const int m = r + half * 8;
        sw[( 0 + ml) * EPAD + m] = acc0[r];
        sw[(16 + ml) * EPAD + m] = acc1[r];
        sw[(32 + ml) * EPAD + m] = acc2[r];
        sw[(48 + ml) * EPAD + m] = acc3[r];
    }
    __syncthreads();

    float* obase = out + b * (O_ * HW_) + h * W_ + wb;
#pragma unroll 8
    for (int i = 0; i < 32; ++i) {
        const int o = i * 2 + half;                // lanes 0-15: even o, 16-31: odd o
        obase[o * HW_ + ml] = sw[o * EPAD + ml] + bias[o];   // 16 consecutive floats/group
    }
}

extern "C" void kernel_launch(void* const* d_in, const int* in_sizes, int n_in,
                              void* d_out, int out_size, void* d_ws, size_t ws_size,
                              hipStream_t stream) {
    (void)in_sizes; (void)n_in; (void)out_size; (void)ws_size;
    const float* x     = (const float*)d_in[0];
    const float* w_off = (const float*)d_in[1];
    const float* b_off = (const float*)d_in[2];
    const float* w     = (const float*)d_in[3];
    const float* bias  = (const float*)d_in[4];
    float* out  = (float*)d_out;
    float* offs = (float*)d_ws;                       // (B,18,H,W) fp32 = 4.5 MB
    float* wtr  = (float*)d_ws + (size_t)B_ * NOFF * HW_;  // interleaved weights, 147 KB

    transpose_w_kernel<<<(KPT * C_ * O_) / 256, 256, 0, stream>>>(w, wtr);
    const int npix = B_ * HW_;                        // 65536
    offset_conv_kernel<<<npix / 256, 256, 0, stream>>>(x, w_off, b_off, offs);
    deform_wmma_kernel<<<npix / 128, 256, 0, stream>>>(x, offs, wtr, bias, out);
}